// MultiHeadSelfAttention_82609400971629
// MI455X (gfx1250) — compile-verified
//
#include <hip/hip_runtime.h>
#include <hip/hip_bf16.h>

typedef __attribute__((ext_vector_type(2))) float v2f;
typedef __attribute__((ext_vector_type(8))) float v8f;

#define BATCH 8
#define SEQ   2048
#define CDIM  1024
#define HS    64
#define BT    (BATCH * SEQ)      // 16384
#define QKVC  192                // 3*hs
#define WAVES 4                  // waves per block

// D = A(16x4) * B(4x16) + C, full fp32 via V_WMMA_F32_16X16X4_F32
static __device__ __forceinline__ v8f wmma4(v2f a, v2f b, v8f c) {
    return __builtin_amdgcn_wmma_f32_16x16x4_f32(
        /*neg_a=*/false, a, /*neg_b=*/false, b,
        /*c_mod=*/(short)0, c, /*reuse_a=*/false, /*reuse_b=*/false);
}

// ---------------------------------------------------------------------------
// Stage 0: fold the 16 identical heads into proj_w:
//   W_eff[j][d] = sum_h proj_w[j][h*64 + d]   (j<1024, d<64)
// ---------------------------------------------------------------------------
__global__ __launch_bounds__(256)
void fold_proj_kernel(const float* __restrict__ proj_w, float* __restrict__ w_eff) {
    int idx = blockIdx.x * blockDim.x + threadIdx.x;   // 65536 threads
    int j = idx >> 6, d = idx & 63;
    float s = 0.f;
#pragma unroll
    for (int h = 0; h < 16; ++h) s += proj_w[(size_t)j * CDIM + h * HS + d];
    w_eff[idx] = s;
}

// ---------------------------------------------------------------------------
// Stage 1: qkv[BT][192] = ids[BT][1024] @ qkv_w[192][1024]^T + qkv_b
// One wave per 32x32 output macro-tile (2x2 wmma tiles): 4 loads -> 4 wmma
// per K-step, K=1024.
// ---------------------------------------------------------------------------
__global__ __launch_bounds__(32 * WAVES)
void qkv_kernel(const float* __restrict__ ids, const float* __restrict__ qkv_w,
                const float* __restrict__ qkv_b, float* __restrict__ qkv) {
    const int lane = threadIdx.x & 31;
    const int wave = threadIdx.x >> 5;
    const int lid  = lane & 15;
    const int half = lane >> 4;
    int task    = blockIdx.x * WAVES + wave;    // (BT/32)*(192/32) = 512*6
    int rowPair = task / 6;
    int colPair = task - rowPair * 6;
    int r0 = rowPair * 32, c0 = colPair * 32;

    const float* a0 = ids   + (size_t)(r0 + lid) * CDIM + 2 * half;
    const float* a1 = a0 + (size_t)16 * CDIM;
    const float* b0 = qkv_w + (size_t)(c0 + lid) * CDIM + 2 * half;
    const float* b1 = b0 + (size_t)16 * CDIM;

    v8f acc[2][2];
#pragma unroll
    for (int i = 0; i < 2; ++i)
#pragma unroll
        for (int j = 0; j < 2; ++j) acc[i][j] = (v8f){};

#pragma unroll 4
    for (int k0 = 0; k0 < CDIM; k0 += 4) {
        v2f A0 = *(const v2f*)(a0 + k0);
        v2f A1 = *(const v2f*)(a1 + k0);
        v2f B0 = *(const v2f*)(b0 + k0);
        v2f B1 = *(const v2f*)(b1 + k0);
        acc[0][0] = wmma4(A0, B0, acc[0][0]);
        acc[0][1] = wmma4(A0, B1, acc[0][1]);
        acc[1][0] = wmma4(A1, B0, acc[1][0]);
        acc[1][1] = wmma4(A1, B1, acc[1][1]);
    }
#pragma unroll
    for (int i = 0; i < 2; ++i)
#pragma unroll
        for (int j = 0; j < 2; ++j) {
            float bias = qkv_b[c0 + 16 * j + lid];
            float* op = qkv + (size_t)(r0 + 16 * i + 8 * half) * QKVC + (c0 + 16 * j + lid);
#pragma unroll
            for (int r = 0; r < 8; ++r) op[(size_t)r * QKVC] = acc[i][j][r] + bias;
        }
}

// ---------------------------------------------------------------------------
// Stage 2: fused causal attention (no softmax):
//   out_h[t][:] = sum_{s<=t} (q[t].k[s]) * v[s]
// One wave per (b, 32-query block): 2 query row-tiles share every K-tile and
// V load. Per 16-key tile: 32 wmma for wei + 32 wmma for out.
// wei round-trips through LDS (C/D layout -> A layout), fenced by s_wait_dscnt.
// ---------------------------------------------------------------------------
__global__ __launch_bounds__(32 * WAVES)
void attn_kernel(const float* __restrict__ qkv, float* __restrict__ out_h) {
    __shared__ float wei_lds[WAVES * 2 * 256];
    const int lane = threadIdx.x & 31;
    const int wave = threadIdx.x >> 5;
    const int lid  = lane & 15;
    const int half = lane >> 4;

    int task = blockIdx.x * WAVES + wave;   // BATCH * (SEQ/32) = 512 tasks
    int b    = task >> 6;
    int qBlk = task & 63;
    int t0   = qBlk * 32;

    // Preload both Q row-tiles in A-matrix layout.
    v2f aq[2][16];
#pragma unroll
    for (int i = 0; i < 2; ++i) {
        const float* qbase = qkv + (size_t)(b * SEQ + t0 + 16 * i + lid) * QKVC + 2 * half;
#pragma unroll
        for (int c = 0; c < 16; ++c) aq[i][c] = *(const v2f*)(qbase + 4 * c);
    }

    v8f dacc[2][4];
#pragma unroll
    for (int i = 0; i < 2; ++i)
#pragma unroll
        for (int n = 0; n < 4; ++n) dacc[i][n] = (v8f){};

    float* wl = &wei_lds[wave * 512];
    const int lastTile = 2 * qBlk + 1;

    for (int sTile = 0; sTile <= lastTile; ++sTile) {
        int s0 = sTile * 16;
        // --- scores: wei_i = Q_i @ K^T, K-tile loads shared by both row tiles ---
        const float* kbase = qkv + (size_t)(b * SEQ + s0 + lid) * QKVC + HS + 2 * half;
        v8f w[2] = {(v8f){}, (v8f){}};
#pragma unroll
        for (int c = 0; c < 16; ++c) {
            v2f bk = *(const v2f*)(kbase + 4 * c);
            w[0] = wmma4(aq[0][c], bk, w[0]);
            w[1] = wmma4(aq[1][c], bk, w[1]);
        }
        // --- causal mask (only top two key-tiles can touch the diagonal) ---
        bool maybeMask = (sTile >= lastTile - 1);
#pragma unroll
        for (int i = 0; i < 2; ++i)
#pragma unroll
            for (int r = 0; r < 8; ++r) {
                int rowLoc = r + 8 * half;
                int rowGlb = t0 + 16 * i + rowLoc;
                float val = (maybeMask && (s0 + lid > rowGlb)) ? 0.f : w[i][r];
                wl[i * 256 + rowLoc * 16 + lid] = val;
            }
        asm volatile("s_wait_dscnt 0" ::: "memory");   // same-wave LDS RAW fence
        // --- accumulate: out_i += wei_i @ V, V loads shared by both row tiles ---
        const float* vbase = qkv + (size_t)(b * SEQ + s0) * QKVC + 2 * HS;
#pragma unroll
        for (int kc = 0; kc < 4; ++kc) {
            v2f aw0 = *(const v2f*)(wl +       lid * 16 + 4 * kc + 2 * half);
            v2f aw1 = *(const v2f*)(wl + 256 + lid * 16 + 4 * kc + 2 * half);
#pragma unroll
            for (int n = 0; n < 4; ++n) {
                const float* vp = vbase + (size_t)(4 * kc + 2 * half) * QKVC + 16 * n + lid;
                v2f bv = { vp[0], vp[QKVC] };
                dacc[0][n] = wmma4(aw0, bv, dacc[0][n]);
                dacc[1][n] = wmma4(aw1, bv, dacc[1][n]);
            }
        }
        asm volatile("" ::: "memory");
    }

#pragma unroll
    for (int i = 0; i < 2; ++i) {
        float* op = out_h + (size_t)(b * SEQ + t0 + 16 * i + 8 * half) * HS + lid;
#pragma unroll
        for (int n = 0; n < 4; ++n)
#pragma unroll
            for (int r = 0; r < 8; ++r)
                op[(size_t)r * HS + 16 * n] = dacc[i][n][r];
    }
}

// ---------------------------------------------------------------------------
// Stage 3: out[BT][1024] = out_h[BT][64] @ W_eff[1024][64]^T + proj_b
// One wave per 32x32 macro-tile; K=64 -> 64 wmma, 64 float2 loads per wave.
// ---------------------------------------------------------------------------
__global__ __launch_bounds__(32 * WAVES)
void proj_kernel(const float* __restrict__ out_h, const float* __restrict__ w_eff,
                 const float* __restrict__ proj_b, float* __restrict__ out) {
    const int lane = threadIdx.x & 31;
    const int wave = threadIdx.x >> 5;
    const int lid  = lane & 15;
    const int half = lane >> 4;

    int task    = blockIdx.x * WAVES + wave;   // (BT/32)*(1024/32) = 512*32
    int rowPair = task >> 5;
    int colPair = task & 31;
    int r0 = rowPair * 32, c0 = colPair * 32;

    const float* a0 = out_h + (size_t)(r0 + lid) * HS + 2 * half;
    const float* a1 = a0 + (size_t)16 * HS;
    const float* b0 = w_eff + (size_t)(c0 + lid) * HS + 2 * half;
    const float* b1 = b0 + (size_t)16 * HS;

    v8f acc[2][2];
#pragma unroll
    for (int i = 0; i < 2; ++i)
#pragma unroll
        for (int j = 0; j < 2; ++j) acc[i][j] = (v8f){};

#pragma unroll 4
    for (int k0 = 0; k0 < HS; k0 += 4) {
        v2f A0 = *(const v2f*)(a0 + k0);
        v2f A1 = *(const v2f*)(a1 + k0);
        v2f B0 = *(const v2f*)(b0 + k0);
        v2f B1 = *(const v2f*)(b1 + k0);
        acc[0][0] = wmma4(A0, B0, acc[0][0]);
        acc[0][1] = wmma4(A0, B1, acc[0][1]);
        acc[1][0] = wmma4(A1, B0, acc[1][0]);
        acc[1][1] = wmma4(A1, B1, acc[1][1]);
    }
#pragma unroll
    for (int i = 0; i < 2; ++i)
#pragma unroll
        for (int j = 0; j < 2; ++j) {
            float bias = proj_b[c0 + 16 * j + lid];
            float* op = out + (size_t)(r0 + 16 * i + 8 * half) * CDIM + (c0 + 16 * j + lid);
#pragma unroll
            for (int r = 0; r < 8; ++r) op[(size_t)r * CDIM] = acc[i][j][r] + bias;
        }
}

// ---------------------------------------------------------------------------
extern "C" void kernel_launch(void* const* d_in, const int* in_sizes, int n_in,
                              void* d_out, int out_size, void* d_ws, size_t ws_size,
                              hipStream_t stream) {
    const float* ids    = (const float*)d_in[0];
    const float* qkv_w  = (const float*)d_in[1];
    const float* qkv_b  = (const float*)d_in[2];
    const float* proj_w = (const float*)d_in[3];
    const float* proj_b = (const float*)d_in[4];
    float* out = (float*)d_out;

    float* qkv  = (float*)d_ws;                        // BT*192 floats (12.6 MB)
    float* outh = qkv  + (size_t)BT * QKVC;            // BT*64  floats (4 MB)
    float* weff = outh + (size_t)BT * HS;              // 1024*64 floats (256 KB)

    fold_proj_kernel<<<CDIM * HS / 256, 256, 0, stream>>>(proj_w, weff);
    qkv_kernel<<<(BT / 32) * (QKVC / 32) / WAVES, 32 * WAVES, 0, stream>>>(ids, qkv_w, qkv_b, qkv);
    attn_kernel<<<BATCH * (SEQ / 32) / WAVES, 32 * WAVES, 0, stream>>>(qkv, outh);
    proj_kernel<<<(BT / 32) * (CDIM / 32) / WAVES, 32 * WAVES, 0, stream>>>(outh, weff, proj_b, out);
}